// AttentionAggregation_14336600834785
// MI455X (gfx1250) — compile-verified
//
#include <hip/hip_runtime.h>
#include <hip/hip_bf16.h>

// ---------------------------------------------------------------------------
// AttentionAggregation: y = segment_sum(alpha[:,None] * (x@W^T)[idx_j], idx_i)
//   x: (N_ATOMS, 128) f32, W: (128,128) f32, alpha/idx_i/idx_j: (N_PAIRS,)
//   idx_i is SORTED -> run-length accumulation before atomic flush.
// Stage 1: v = x @ W^T via V_WMMA_F32_16X16X4_F32 (v kept in d_ws, 25.6 MB)
// Stage 2: zero y
// Stage 3: wave-per-chunk gather (float4/lane) + run-length scatter-add
// ---------------------------------------------------------------------------

typedef float v2f __attribute__((ext_vector_type(2)));
typedef float v8f __attribute__((ext_vector_type(8)));

#define FDIM 128
#define TILE 16
#define PAIRS_PER_WAVE 32

// ---------------------------------------------------------------------------
// Stage 1: v = x @ W^T.  One block = 16 rows of x (staged in LDS, 8 KB).
// 8 waves per block each own one 16-wide N-tile (8*16 = 128 = FDIM).
// Per wave: 32x V_WMMA_F32_16X16X4_F32 accumulating K=128.
// ---------------------------------------------------------------------------
__global__ __launch_bounds__(256) void gemm_xwt_wmma(
    const float* __restrict__ x, const float* __restrict__ W,
    float* __restrict__ v, int n_atoms)
{
    __shared__ float xs[TILE * FDIM];   // 16x128 f32 slab = 8 KB

    const int wave = threadIdx.x >> 5;
    const int lane = threadIdx.x & 31;
    const int m0   = blockIdx.x * TILE;

    // Cooperative load of the 16x128 x-slab into LDS (float4 per access).
    // FDIM/4 = 32 float4 per row -> row index = t >> 5.
    {
        float4* dst = (float4*)xs;
        const int n4 = TILE * FDIM / 4;             // 512 float4
        for (int t = threadIdx.x; t < n4; t += blockDim.x) {
            const int row = t >> 5;                  // which of the 16 rows
            float4 val = make_float4(0.f, 0.f, 0.f, 0.f);
            if (m0 + row < n_atoms) {
                val = ((const float4*)(x + (size_t)(m0 + row) * FDIM))[t & 31];
            }
            dst[t] = val;
        }
    }
    __syncthreads();

    // A-matrix 16x4 f32 layout: lanes 0-15 -> M=0..15 ; VGPR0 holds K=0 (lanes
    // 0-15) / K=2 (lanes 16-31), VGPR1 holds K=1 / K=3.
    // B-matrix 4x16: lanes 0-15 -> N=0..15, same K striping.
    const int n0   = wave * TILE;                   // this wave's N-tile
    const int mrow = lane & 15;
    const int ncol = lane & 15;
    const int koff = (lane >> 4) << 1;              // 0 or 2

    const float* Wrow = W + (size_t)(n0 + ncol) * FDIM;  // B[k][n] = W[n][k]

    v8f acc = {};
    #pragma unroll
    for (int k = 0; k < FDIM; k += 4) {
        v2f a, b;
        a.x = xs[mrow * FDIM + k + koff];
        a.y = xs[mrow * FDIM + k + koff + 1];
        b.x = Wrow[k + koff];
        b.y = Wrow[k + koff + 1];
        // (neg_a, A, neg_b, B, c_mod, C, reuse_a, reuse_b)
        acc = __builtin_amdgcn_wmma_f32_16x16x4_f32(
            false, a, false, b, (short)0, acc, false, false);
    }

    // C/D 16x16 f32 layout: VGPR r -> row r (lanes 0-15) / row r+8 (lanes 16-31)
    const int mbase = m0 + ((lane >> 4) << 3);
    #pragma unroll
    for (int r = 0; r < 8; ++r) {
        const int m = mbase + r;
        if (m < n_atoms) v[(size_t)m * FDIM + n0 + ncol] = acc[r];
    }
}

// ---------------------------------------------------------------------------
// Stage 2: zero the output (harness poisons d_out with 0xAA).
// ---------------------------------------------------------------------------
__global__ __launch_bounds__(256) void zero_f32(float* __restrict__ p, int n)
{
    const int i4 = (blockIdx.x * blockDim.x + threadIdx.x) * 4;
    if (i4 + 3 < n) {
        *(float4*)(p + i4) = make_float4(0.f, 0.f, 0.f, 0.f);
    } else {
        for (int t = i4; t < n; ++t) p[t] = 0.f;
    }
}

// ---------------------------------------------------------------------------
// Stage 3: gather + scale + run-length segmented scatter-add.
// One wave handles PAIRS_PER_WAVE consecutive pairs; lane l owns features
// [4l, 4l+4).  Accumulate in registers while idx_i is unchanged (it's sorted,
// avg run = 32), flush with global_atomic_add_f32 on segment change.
// ---------------------------------------------------------------------------
__global__ __launch_bounds__(256) void scatter_segsum(
    const float* __restrict__ v, const float* __restrict__ alpha,
    const int* __restrict__ idx_i, const int* __restrict__ idx_j,
    float* __restrict__ y, int n_pairs)
{
    const int gwave = blockIdx.x * (blockDim.x >> 5) + (threadIdx.x >> 5);
    const int lane  = threadIdx.x & 31;

    const int p0 = gwave * PAIRS_PER_WAVE;
    if (p0 >= n_pairs) return;
    const int p1 = min(p0 + PAIRS_PER_WAVE, n_pairs);

    const int foff = lane * 4;                      // 32 lanes * float4 = 128
    float4 acc = make_float4(0.f, 0.f, 0.f, 0.f);
    int cur = -1;

    for (int p = p0; p < p1; ++p) {
        const int   i = idx_i[p];
        const int   j = idx_j[p];
        const float a = alpha[p];

        // Prefetch next pair's gathered row (global_prefetch_b8).
        if (p + 1 < p1) {
            __builtin_prefetch(v + (size_t)idx_j[p + 1] * FDIM + foff, 0, 0);
        }

        if (i != cur) {
            if (cur >= 0) {
                float* dst = y + (size_t)cur * FDIM + foff;
                atomicAdd(dst + 0, acc.x);
                atomicAdd(dst + 1, acc.y);
                atomicAdd(dst + 2, acc.z);
                atomicAdd(dst + 3, acc.w);
            }
            acc = make_float4(0.f, 0.f, 0.f, 0.f);
            cur = i;
        }

        const float4 vj = *(const float4*)(v + (size_t)j * FDIM + foff);
        acc.x += a * vj.x;
        acc.y += a * vj.y;
        acc.z += a * vj.z;
        acc.w += a * vj.w;
    }

    if (cur >= 0) {
        float* dst = y + (size_t)cur * FDIM + foff;
        atomicAdd(dst + 0, acc.x);
        atomicAdd(dst + 1, acc.y);
        atomicAdd(dst + 2, acc.z);
        atomicAdd(dst + 3, acc.w);
    }
}

// ---------------------------------------------------------------------------
extern "C" void kernel_launch(void* const* d_in, const int* in_sizes, int n_in,
                              void* d_out, int out_size, void* d_ws, size_t ws_size,
                              hipStream_t stream)
{
    const float* x     = (const float*)d_in[0];
    const float* alpha = (const float*)d_in[1];
    const int*   idx_i = (const int*)d_in[2];
    const int*   idx_j = (const int*)d_in[3];
    const float* W     = (const float*)d_in[4];
    float*       y     = (float*)d_out;
    float*       v     = (float*)d_ws;          // (n_atoms, 128) f32 = 25.6 MB

    const int n_atoms = in_sizes[0] / FDIM;
    const int n_pairs = in_sizes[1];

    // Stage 1: v = x @ W^T  (one block per 16-row slab; 8 waves cover N=128)
    const int mtiles = (n_atoms + TILE - 1) / TILE;
    gemm_xwt_wmma<<<mtiles, 256, 0, stream>>>(x, W, v, n_atoms);

    // Stage 2: y = 0
    const int zblocks = ((out_size + 3) / 4 + 255) / 256;
    zero_f32<<<zblocks, 256, 0, stream>>>(y, out_size);

    // Stage 3: gather/scale/segmented scatter-add
    const int waves  = (n_pairs + PAIRS_PER_WAVE - 1) / PAIRS_PER_WAVE;
    const int blocks = (waves + 7) / 8;             // 8 waves per 256-thr block
    scatter_segsum<<<blocks, 256, 0, stream>>>(v, alpha, idx_i, idx_j, y, n_pairs);
}